// GCNLayer_66365834658161
// MI455X (gfx1250) — compile-verified
//
#include <hip/hip_runtime.h>
#include <hip/hip_bf16.h>
#include <stdint.h>

#define D64 64
#define BN_EPS 1e-5f
#define DROP_P 0.1f

typedef float v2f __attribute__((ext_vector_type(2)));
typedef float v8f __attribute__((ext_vector_type(8)));

// ---------------------------------------------------------------------------
// Kernel 1: init deg[i] = 1.0 (self loop) and zero the 128-float BN stats
// ---------------------------------------------------------------------------
__global__ void k_init(float* __restrict__ deg, float* __restrict__ stats, int n) {
    int i = blockIdx.x * blockDim.x + threadIdx.x;
    if (i < n)   deg[i] = 1.0f;
    if (i < 128) stats[i] = 0.0f;
}

// ---------------------------------------------------------------------------
// Kernel 2: degree accumulation over edge destinations
// ---------------------------------------------------------------------------
__global__ void k_degree(const int* __restrict__ ei, float* __restrict__ deg, int e) {
    int i = blockIdx.x * blockDim.x + threadIdx.x;
    if (i >= e) return;
    int dst = ei[(size_t)e + i];
    atomicAdd(&deg[dst], 1.0f);
}

// ---------------------------------------------------------------------------
// Kernel 3: deg -> dinv = rsqrt(max(deg,1)) in place
// ---------------------------------------------------------------------------
__global__ void k_dinv(float* __restrict__ deg, int n) {
    int i = blockIdx.x * blockDim.x + threadIdx.x;
    if (i >= n) return;
    deg[i] = __frsqrt_rn(fmaxf(deg[i], 1.0f));
}

// ---------------------------------------------------------------------------
// Kernel 4: h = x @ W via V_WMMA_F32_16X16X4_F32.
// One wave per 16-row tile; 4 output tiles of 16 cols; K=64 in steps of 4.
// A frag (16x4 f32, 2 VGPRs): lanes 0-15 hold K={0,1}, lanes 16-31 K={2,3}.
// B frag (4x16 f32, 2 VGPRs): mirrored striping across lanes.
// C/D (16x16 f32, 8 VGPRs): VGPR j = row j (lanes 0-15) / row j+8 (lanes 16-31).
// ---------------------------------------------------------------------------
__global__ void k_gemm_wmma(const float* __restrict__ x, const float* __restrict__ W,
                            float* __restrict__ h, int nTiles) {
    int wave = (int)((blockIdx.x * blockDim.x + threadIdx.x) >> 5);
    if (wave >= nTiles) return;             // uniform per wave: EXEC stays all-1s
    int lane = threadIdx.x & 31;
    int half = lane >> 4;                   // 0 or 1
    int r    = lane & 15;
    int m0   = wave * 16;

    const float* xrow = x + (size_t)(m0 + r) * D64;

    v8f acc0 = {}, acc1 = {}, acc2 = {}, acc3 = {};

    #pragma unroll
    for (int kk = 0; kk < D64; kk += 4) {
        int ka = kk + 2 * half;
        v2f a;
        a.x = xrow[ka];
        a.y = xrow[ka + 1];

        const float* w0 = W + (size_t)ka * D64 + r;
        const float* w1 = w0 + D64;         // next K row of W
        v2f b0, b1, b2, b3;
        b0.x = w0[0];  b0.y = w1[0];
        b1.x = w0[16]; b1.y = w1[16];
        b2.x = w0[32]; b2.y = w1[32];
        b3.x = w0[48]; b3.y = w1[48];

        acc0 = __builtin_amdgcn_wmma_f32_16x16x4_f32(false, a, false, b0, (short)0, acc0, false, false);
        acc1 = __builtin_amdgcn_wmma_f32_16x16x4_f32(false, a, false, b1, (short)0, acc1, false, false);
        acc2 = __builtin_amdgcn_wmma_f32_16x16x4_f32(false, a, false, b2, (short)0, acc2, false, false);
        acc3 = __builtin_amdgcn_wmma_f32_16x16x4_f32(false, a, false, b3, (short)0, acc3, false, false);
    }

    #pragma unroll
    for (int j = 0; j < 8; ++j) {
        size_t row = (size_t)(m0 + j + 8 * half);
        float* hp = h + row * D64 + r;
        hp[0]  = acc0[j];
        hp[16] = acc1[j];
        hp[32] = acc2[j];
        hp[48] = acc3[j];
    }
}

// ---------------------------------------------------------------------------
// Kernel 5: out = h * dinv^2 (self loop) + bias  (full overwrite of d_out)
// ---------------------------------------------------------------------------
__global__ void k_selfloop(const float* __restrict__ h, const float* __restrict__ dinv,
                           const float* __restrict__ bias, float* __restrict__ out,
                           long long total) {
    long long idx = (long long)blockIdx.x * blockDim.x + threadIdx.x;
    if (idx >= total) return;
    int node = (int)(idx >> 6);
    int c    = (int)(idx & 63);
    float di = dinv[node];
    out[idx] = h[idx] * di * di + bias[c];
}

// ---------------------------------------------------------------------------
// Kernel 6: edge aggregation. One wave per edge; each lane owns 2 columns.
// out[dst] += h[src] * dinv[src] * dinv[dst]  via f32 atomics (resolve in L2).
// ---------------------------------------------------------------------------
__global__ void k_edges(const int* __restrict__ ei, const float* __restrict__ h,
                        const float* __restrict__ dinv, float* __restrict__ out, int e) {
    int wave = (int)((blockIdx.x * blockDim.x + threadIdx.x) >> 5);
    if (wave >= e) return;
    int lane = threadIdx.x & 31;
    int s = ei[wave];
    int d = ei[(size_t)e + wave];
    float nrm = dinv[s] * dinv[d];
    const float* hs = h + (size_t)s * D64 + lane * 2;
    float v0 = hs[0] * nrm;
    float v1 = hs[1] * nrm;
    float* op = out + (size_t)d * D64 + lane * 2;
    atomicAdd(op,     v0);
    atomicAdd(op + 1, v1);
}

// ---------------------------------------------------------------------------
// Kernel 7: BN column stats (sum, sumsq) -> stats[0:64], stats[64:128]
// ---------------------------------------------------------------------------
__global__ void k_bnstats(const float* __restrict__ out, float* __restrict__ stats, int n) {
    __shared__ float ssum[256];
    __shared__ float ssq[256];
    int c  = threadIdx.x & 63;
    int rl = threadIdx.x >> 6;                 // 0..3
    float s = 0.0f, q = 0.0f;
    for (int row = blockIdx.x * 4 + rl; row < n; row += gridDim.x * 4) {
        float v = out[(size_t)row * D64 + c];
        s += v;
        q += v * v;
    }
    ssum[threadIdx.x] = s;
    ssq[threadIdx.x]  = q;
    __syncthreads();
    if (threadIdx.x < 64) {
        float S = ssum[c] + ssum[c + 64] + ssum[c + 128] + ssum[c + 192];
        float Q = ssq[c]  + ssq[c + 64]  + ssq[c + 128]  + ssq[c + 192];
        atomicAdd(&stats[c],      S);
        atomicAdd(&stats[64 + c], Q);
    }
}

// ---------------------------------------------------------------------------
// Kernel 8: BN normalize + affine, hash dropout (p=0.1), ReLU — in place.
// ---------------------------------------------------------------------------
__global__ void k_finalize(float* __restrict__ out, const float* __restrict__ stats,
                           const float* __restrict__ gamma, const float* __restrict__ beta,
                           int n, long long total) {
    long long idx = (long long)blockIdx.x * blockDim.x + threadIdx.x;
    if (idx >= total) return;
    int c = (int)(idx & 63);
    float invN = 1.0f / (float)n;
    float mean = stats[c] * invN;
    float var  = stats[64 + c] * invN - mean * mean;
    float y = (out[idx] - mean) * __frsqrt_rn(var + BN_EPS) * gamma[c] + beta[c];

    // deterministic hash "bernoulli(0.9)" stand-in for jax threefry(key=42)
    uint32_t z = (uint32_t)idx + 0x9E3779B9u * 42u;
    z ^= z >> 16; z *= 0x85EBCA6Bu;
    z ^= z >> 13; z *= 0xC2B2AE35u;
    z ^= z >> 16;
    float u = (float)(z >> 8) * (1.0f / 16777216.0f);
    y = (u < (1.0f - DROP_P)) ? y * (1.0f / (1.0f - DROP_P)) : 0.0f;

    out[idx] = fmaxf(y, 0.0f);
}

// ---------------------------------------------------------------------------
// Host-side launcher
// ---------------------------------------------------------------------------
extern "C" void kernel_launch(void* const* d_in, const int* in_sizes, int n_in,
                              void* d_out, int out_size, void* d_ws, size_t ws_size,
                              hipStream_t stream) {
    const float* x     = (const float*)d_in[0];
    const int*   ei    = (const int*)d_in[1];     // [2, E] edge index
    const float* W     = (const float*)d_in[2];
    const float* bias  = (const float*)d_in[3];
    const float* gamma = (const float*)d_in[4];
    const float* beta  = (const float*)d_in[5];
    float*       out   = (float*)d_out;

    const int n = in_sizes[0] / D64;              // 100000
    const int e = in_sizes[1] / 2;                // 1250000
    const long long total = (long long)n * D64;

    // workspace layout: h [n*64] | deg/dinv [n] | stats [128]
    float* h_ws  = (float*)d_ws;
    float* deg   = h_ws + (size_t)n * D64;
    float* stats = deg + n;

    // 1. init deg + stats
    k_init<<<(n + 255) / 256, 256, 0, stream>>>(deg, stats, n);
    // 2. degree over dst
    k_degree<<<(e + 255) / 256, 256, 0, stream>>>(ei, deg, e);
    // 3. dinv
    k_dinv<<<(n + 255) / 256, 256, 0, stream>>>(deg, n);
    // 4. h = x @ W (WMMA): one wave per 16-row tile, 8 waves per block
    int nTiles = (n + 15) / 16;
    k_gemm_wmma<<<(nTiles + 7) / 8, 256, 0, stream>>>(x, W, h_ws, nTiles);
    // 5. self-loop term + bias (overwrites d_out)
    k_selfloop<<<(int)((total + 255) / 256), 256, 0, stream>>>(h_ws, deg, bias, out, total);
    // 6. edge scatter-add: one wave per edge
    k_edges<<<(e + 7) / 8, 256, 0, stream>>>(ei, h_ws, deg, out, e);
    // 7. BN stats
    k_bnstats<<<512, 256, 0, stream>>>(out, stats, n);
    // 8. normalize + dropout + relu
    k_finalize<<<(int)((total + 255) / 256), 256, 0, stream>>>(out, stats, gamma, beta, n, total);
}